// RoiPoolingConv_14817637171644
// MI455X (gfx1250) — compile-verified
//
#include <hip/hip_runtime.h>

// ROI bilinear pooling, MI455X (gfx1250).
// img  : (1, 128, 128, 512) f32 NHWC
// rois : (1, 300, 4) i32  (x, y, w, h)
// out  : (1, 300, 14, 14, 512) f32
//
// Bandwidth-bound gather kernel (~153 MB HBM traffic -> ~6.6us floor at
// 23.3 TB/s; FLOPs negligible, so WMMA is inapplicable). CDNA5 plan:
//  * async global->LDS engine (ASYNCcnt) double-buffers the next px's four
//    corner rows while the VALU blends the current px from LDS,
//  * per-wave s_wait_asynccnt only (each lane owns its 16B slot -> no
//    barriers; async loads retire in order),
//  * non-temporal b128 stores keep the 120MB output stream from evicting
//    the L2-resident 32MB image.

typedef float v4f __attribute__((ext_vector_type(4)));

#define IMG_W 128
#define IMG_C 512
#define N_ROI 300
#define POOL  14
#define NTHR  128   // 4 x wave32; one float4 channel slice per lane

__device__ __forceinline__ void axis_coords(int start, int size, float scale,
                                            int p, int& a0, int& a1, float& f) {
    float sf  = (float)size;
    float src = ((float)p + 0.5f) * scale - 0.5f;   // scale = size/POOL (hoisted)
    src = fminf(fmaxf(src, 0.0f), sf - 1.0f);
    int i0 = (int)src;              // src >= 0 -> trunc == floor
    f = src - (float)i0;
    int i1 = min(i0 + 1, size - 1);
    a0 = start + i0;
    a1 = start + i1;
}

// Fire-and-forget DMA of 16B/lane into LDS; tracked by ASYNCcnt.
__device__ __forceinline__ void async_ld_b128(unsigned int lds_addr, const float* gptr) {
    unsigned long long ga = (unsigned long long)(uintptr_t)gptr; // flat == global VA
    asm volatile("global_load_async_to_lds_b128 %0, %1, off"
                 :: "v"(lds_addr), "v"(ga)
                 : "memory");
}

__device__ __forceinline__ void stage_corners(unsigned int lbase, unsigned int cstride,
                                              const float* row0, const float* row1,
                                              int x0, int x1, int c) {
    async_ld_b128(lbase + 0u * cstride, row0 + (size_t)x0 * IMG_C + c);
    async_ld_b128(lbase + 1u * cstride, row0 + (size_t)x1 * IMG_C + c);
    async_ld_b128(lbase + 2u * cstride, row1 + (size_t)x0 * IMG_C + c);
    async_ld_b128(lbase + 3u * cstride, row1 + (size_t)x1 * IMG_C + c);
}

__global__ __launch_bounds__(NTHR)
void roi_pool_rows(const float* __restrict__ img,
                   const int*   __restrict__ rois,
                   float*       __restrict__ out) {
    __shared__ v4f sbuf[2][4][NTHR];            // double buffer: 2 x 4 corners x 2KB

    const int blk = blockIdx.x;                 // r * POOL + py
    const int r   = blk / POOL;
    const int py  = blk - r * POOL;
    const int t   = threadIdx.x;
    const int c   = t << 2;                     // channel offset (float4 granularity)

    // Block-uniform ROI -> s_load.
    const int rx = rois[r * 4 + 0];
    const int ry = rois[r * 4 + 1];
    const int rw = rois[r * 4 + 2];
    const int rh = rois[r * 4 + 3];

    // Hoist the two FP divides (v_div_scale/v_rcp chain) out of the px loop.
    const float xscale = (float)rw / (float)POOL;
    const float yscale = (float)rh / (float)POOL;

    int y0, y1; float fy;
    axis_coords(ry, rh, yscale, py, y0, y1, fy);

    const float* row0 = img + (size_t)y0 * (IMG_W * IMG_C);
    const float* row1 = img + (size_t)y1 * (IMG_W * IMG_C);
    float*       orow = out + (size_t)blk * (POOL * IMG_C) + c;

    // LDS byte addresses of this lane's slots (flat-LDS ptr: offset = low 32b).
    unsigned int lbase0  = (unsigned int)(uintptr_t)&sbuf[0][0][t];
    unsigned int lbase1  = (unsigned int)(uintptr_t)&sbuf[1][0][t];
    const unsigned int cstride = NTHR * sizeof(v4f);   // 2048B per corner plane

    int x0, x1; float fx;
    axis_coords(rx, rw, xscale, 0, x0, x1, fx);
    stage_corners(lbase0, cstride, row0, row1, x0, x1, c);   // preload px = 0

    for (int px = 0; px < POOL; ++px) {
        const int b = px & 1;
        const float cfx = fx;                   // weights for the buffer we consume

        if (px + 1 < POOL) {
            // Stage px+1 into the other buffer, then wait for the *current*
            // buffer's 4 in-order async loads (<=4 outstanding => first 4 done).
            int nx0, nx1; float nfx;
            axis_coords(rx, rw, xscale, px + 1, nx0, nx1, nfx);
            stage_corners(b ? lbase0 : lbase1, cstride, row0, row1, nx0, nx1, c);
            x0 = nx0; x1 = nx1; fx = nfx;
            asm volatile("s_wait_asynccnt 0x4" ::: "memory");
        } else {
            asm volatile("s_wait_asynccnt 0x0" ::: "memory");
        }

        // Consume this lane's slots (ds_load_b128 x4) and blend.
        v4f c00 = sbuf[b][0][t];
        v4f c01 = sbuf[b][1][t];
        v4f c10 = sbuf[b][2][t];
        v4f c11 = sbuf[b][3][t];

        v4f top = c00 + (c01 - c00) * cfx;
        v4f bot = c10 + (c11 - c10) * cfx;
        v4f res = top + (bot - top) * fy;

        // Streaming output: non-temporal b128 store (th:TH_STORE_NT).
        __builtin_nontemporal_store(res, (v4f*)(orow + (size_t)px * IMG_C));
    }
}

extern "C" void kernel_launch(void* const* d_in, const int* in_sizes, int n_in,
                              void* d_out, int out_size, void* d_ws, size_t ws_size,
                              hipStream_t stream) {
    const float* img  = (const float*)d_in[0];
    const int*   rois = (const int*)d_in[1];
    // d_in[2] is pool_size scalar; POOL=14 is baked in at compile time.
    float* out = (float*)d_out;

    dim3 grid(N_ROI * POOL);   // 4200 blocks: one (roi, pooled_row) each
    dim3 block(NTHR);
    roi_pool_rows<<<grid, block, 0, stream>>>(img, rois, out);
}